// Correlation_5205500362858
// MI455X (gfx1250) — compile-verified
//
#include <hip/hip_runtime.h>
#include <hip/hip_bf16.h>

#define BB 8
#define HH 384
#define WW 512
#define CC 32
#define SS 32
#define TPB 256   // threads per block = pixels per block

typedef __attribute__((ext_vector_type(4))) unsigned int v4u;
typedef __attribute__((ext_vector_type(8))) unsigned int v8u;

__global__ __launch_bounds__(TPB) void
Correlation_5205500362858_kernel(const float* __restrict__ cntr,
                                 const float* __restrict__ srnd,
                                 const float* __restrict__ scale,
                                 const float* __restrict__ trans_x,
                                 const float* __restrict__ trans_y,
                                 float* __restrict__ out)
{
    // LDS staging buffer for this block's cntr slice: 256 pixels * 32 ch * 4B = 32 KB
    __shared__ float smem[TPB * CC];

    const int tid = threadIdx.x;
    const int x0  = blockIdx.x * TPB;
    const int y   = blockIdx.y;
    const int b   = blockIdx.z;

    // ---- TDM: DMA cntr[b, y, x0:x0+256, :] (8192 contiguous floats) into LDS ----
    if (tid < 32) {  // wave 0 only issues the tensor DMA (EXEC ignored by TDM)
        unsigned long long ga =
            (unsigned long long)(const void*)(cntr + (((size_t)b * HH + y) * WW + x0) * CC);
        unsigned lds_off = (unsigned)(unsigned long long)(void*)smem; // low 32 bits = LDS byte offset

        v4u g0;
        g0.x = 1u;                                   // count=1 valid user descriptor
        g0.y = lds_off;                              // lds_addr
        g0.z = (unsigned)ga;                         // global_addr[31:0]
        g0.w = (unsigned)(ga >> 32) | 0x80000000u;   // global_addr[56:32] | type=2 ("image")

        v8u g1;
        g1[0] = 2u << 16;                 // data_size = 2 (4 bytes); no pad/iterate/multicast
        g1[1] = 0x2000u << 16;            // tensor_dim0 = 8192 (lo16 at [31:16])
        g1[2] = 1u << 16;                 // tensor_dim0 hi16 = 0 | tensor_dim1 = 1 (lo16)
        g1[3] = 0x2000u << 16;            // tensor_dim1 hi16 = 0 | tile_dim0 = 8192
        g1[4] = 0u;                       // tile_dim1 = 0 (unused), tile_dim2 = 0 -> 1-D tile
        g1[5] = 0x2000u;                  // tensor_dim0_stride = 8192 (lo32)
        g1[6] = 0u;                       // stride hi16 | tensor_dim1_stride lo16
        g1[7] = 0u;

        // Two-operand form: VADDR2/VADDR3 = NULL (tensor up to 2D)
        asm volatile("tensor_load_to_lds %0, %1" :: "s"(g0), "s"(g1) : "memory");
        __builtin_amdgcn_s_wait_tensorcnt(0);
    }
    __syncthreads();

    // ---- Pull this thread's cntr vector (32 floats) from LDS into registers ----
    const float4* cn4 = (const float4*)(smem) + (size_t)tid * (CC / 4);
    float4 c[8];
#pragma unroll
    for (int k = 0; k < 8; ++k) c[k] = cn4[k];

    const int   x  = x0 + tid;
    const float fx = (float)x;
    const float fy = (float)y;

    const float* sb  = scale   + b * SS;
    const float* txb = trans_x + b * SS;
    const float* tyb = trans_y + b * SS;

    // Uniform 64-bit bases; per-lane offsets kept in 32 bits -> GVS addressing.
    const char* srb = (const char*)(srnd + (size_t)b * HH * WW * CC);
    char*       obp = (char*)out;
    // out byte offset for this pixel: pixel_index * STEPS * 4 = pixel * 128 (max ~201MB < 2^31)
    const unsigned ob_off = (unsigned)(((b * HH + y) * WW + x)) * (SS * 4u);

    // ---- 32 warp steps, 4 at a time so results store as one b128 ----
    for (int s4 = 0; s4 < SS; s4 += 4) {
        float4 res;
#pragma unroll
        for (int j = 0; j < 4; ++j) {
            const int s  = s4 + j;
            const float sc = sb[s];
            const float tX = txb[s];
            const float tY = tyb[s];

            // TF nearest: idx = floor(q + 0.5)
            const float xq = __builtin_fmaf(sc, fx, tX);
            const float yq = __builtin_fmaf(sc, fy, tY);
            const int   xi = (int)__builtin_floorf(xq + 0.5f);
            const int   yi = (int)__builtin_floorf(yq + 0.5f);

            const bool valid = ((unsigned)xi < (unsigned)WW) & ((unsigned)yi < (unsigned)HH);
            const int  xc = min(max(xi, 0), WW - 1);
            const int  yc = min(max(yi, 0), HH - 1);

            // srnd byte offset: yc * (W*C*4 = 65536)  |  xc * (C*4 = 128); disjoint bit fields.
            const unsigned goff = ((unsigned)yc << 16) | ((unsigned)xc << 7);
            const float4* g4 = (const float4*)(srb + goff);

            float a0 = 0.f, a1 = 0.f, a2 = 0.f, a3 = 0.f;
#pragma unroll
            for (int k = 0; k < 8; ++k) {
                const float4 g = g4[k];
                a0 = __builtin_fmaf(c[k].x, g.x, a0);
                a1 = __builtin_fmaf(c[k].y, g.y, a1);
                a2 = __builtin_fmaf(c[k].z, g.z, a2);
                a3 = __builtin_fmaf(c[k].w, g.w, a3);
            }
            const float r = (a0 + a1) + (a2 + a3);
            (&res.x)[j] = valid ? r : 0.0f;
        }
        *(float4*)(obp + ob_off + (unsigned)s4 * 4u) = res;
    }
}

extern "C" void kernel_launch(void* const* d_in, const int* in_sizes, int n_in,
                              void* d_out, int out_size, void* d_ws, size_t ws_size,
                              hipStream_t stream) {
    const float* cntr    = (const float*)d_in[0];
    const float* srnd    = (const float*)d_in[1];
    const float* scale   = (const float*)d_in[2];
    const float* trans_x = (const float*)d_in[3];
    const float* trans_y = (const float*)d_in[4];
    float* out = (float*)d_out;

    dim3 grid(WW / TPB, HH, BB);  // (2, 384, 8)
    Correlation_5205500362858_kernel<<<grid, TPB, 0, stream>>>(
        cntr, srnd, scale, trans_x, trans_y, out);
}